// EdgeToEdgeAggregation_188978561191
// MI455X (gfx1250) — compile-verified
//
#include <hip/hip_runtime.h>
#include <math.h>

// ---------------- problem constants ----------------
#define IN_CH   16
#define HEADS   4
#define OUT_CH  128
#define HC      (HEADS * OUT_CH)   // 512
#define NEG_SLOPE 0.2f

typedef __attribute__((ext_vector_type(2))) float v2f;
typedef __attribute__((ext_vector_type(8))) float v8f;

// ---------------- helpers ----------------
__device__ __forceinline__ void atomicMaxFloat(float* addr, float val) {
  // order-preserving float max via integer atomics (works for mixed signs,
  // init value must be -inf). Positive floats compare as ints; negative
  // floats compare reversed as uints.
  if (val >= 0.0f)
    atomicMax((int*)addr, __float_as_int(val));
  else
    atomicMin((unsigned int*)addr, __float_as_uint(val));
}

__device__ __forceinline__ void atomAddF32(float* addr, float v) {
  __hip_atomic_fetch_add(addr, v, __ATOMIC_RELAXED, __HIP_MEMORY_SCOPE_AGENT);
}

__device__ __forceinline__ float leaky(float x) {
  return x > 0.0f ? x : NEG_SLOPE * x;
}

// ---------------- kernel 1: init out=bias, smax=-inf, denom=0 ----------------
__global__ void gat_init(float* __restrict__ out, const float* __restrict__ bias,
                         float* __restrict__ smax, float* __restrict__ denom,
                         int n_nodes) {
  int i = blockIdx.x * blockDim.x + threadIdx.x;
  int total = n_nodes * HC;
  if (i < total) out[i] = bias[i & (HC - 1)];
  if (i < n_nodes * HEADS) { smax[i] = -INFINITY; denom[i] = 0.0f; }
}

// ---------------- kernel 2: x = edge_feat @ W via V_WMMA_F32_16X16X4_F32 ----
// One wave computes a 16(M)x16(N) f32 tile, K=16 via 4 chained WMMAs.
// A (16x4 f32): lanes 0-15 hold row M=lane; VGPR0 = K base+{0 | 2}, VGPR1 = +1/+3
//   split on lane half (per ISA 7.12.2 "32-bit A-Matrix 16x4").
// B (4x16 f32): row-striped across lanes symmetrically.
// D (16x16 f32, 8 VGPRs): VGPR v -> M = v (+8 for lanes 16-31), N = lane%16.
__global__ void gat_gemm_wmma(const float* __restrict__ ef,
                              const float* __restrict__ Wl,
                              const float* __restrict__ Wr,
                              float* __restrict__ xl,
                              float* __restrict__ xr) {
  const int wave = threadIdx.x >> 5;
  const int lane = threadIdx.x & 31;
  const int mt = blockIdx.x;                  // M tile: 0..(N_NODES/16 - 1)
  const int nt = (blockIdx.y << 2) + wave;    // N tile: 0..31
  const float* __restrict__ W = blockIdx.z ? Wr : Wl;
  float* __restrict__ X = blockIdx.z ? xr : xl;

  const int m0 = mt * 16;
  const int n0 = nt * 16;
  const int half = lane >> 4;                 // 0: lanes 0-15, 1: lanes 16-31
  const int lid  = lane & 15;

  const float* __restrict__ arow = ef + (m0 + lid) * IN_CH + 2 * half;
  const float* __restrict__ bcol = W + (2 * half) * HC + n0 + lid;

  v8f acc = {};
#pragma unroll
  for (int kk = 0; kk < 4; ++kk) {           // K = kk*4 .. kk*4+3
    v2f a, b;
    a.x = arow[kk * 4 + 0];                  // A[M=lid][k0 + 2*half]
    a.y = arow[kk * 4 + 1];                  // A[M=lid][k0 + 2*half + 1]
    b.x = bcol[(kk * 4)     * HC];           // B[k0 + 2*half    ][N=lid]
    b.y = bcol[(kk * 4 + 1) * HC];           // B[k0 + 2*half + 1][N=lid]
    acc = __builtin_amdgcn_wmma_f32_16x16x4_f32(
        /*neg_a=*/false, a, /*neg_b=*/false, b,
        /*c_mod=*/(short)0, acc, /*reuse_a=*/false, /*reuse_b=*/false);
  }

  float* __restrict__ drow = X + (m0 + half * 8) * HC + n0 + lid;
#pragma unroll
  for (int v = 0; v < 8; ++v) drow[v * HC] = acc[v];
}

// ---------------- kernel 3: per-edge score + segment max --------------------
// One wave per edge (incl. self loops). Per head: 32 lanes x float4 over the
// 128 channels, leaky(xl+xr)*att, shfl_xor wave reduction.
__global__ void gat_score_max(const long long* __restrict__ ei,
                              const float* __restrict__ xl,
                              const float* __restrict__ xr,
                              const float* __restrict__ att,
                              float* __restrict__ score,
                              float* __restrict__ smax,
                              int n_edges, int n_tot) {
  const int wid  = (blockIdx.x * blockDim.x + threadIdx.x) >> 5;
  const int lane = threadIdx.x & 31;
  if (wid >= n_tot) return;

  long long s, d;
  if (wid < n_edges) { s = ei[wid]; d = ei[n_edges + wid]; }
  else               { s = d = (long long)(wid - n_edges); }

  const float* __restrict__ pl = xl + s * HC;
  const float* __restrict__ pr = xr + d * HC;

#pragma unroll
  for (int h = 0; h < HEADS; ++h) {
    const int off = h * OUT_CH + lane * 4;
    float4 a = *(const float4*)(pl + off);
    float4 b = *(const float4*)(pr + off);
    float4 w = *(const float4*)(att + off);
    float sum = leaky(a.x + b.x) * w.x + leaky(a.y + b.y) * w.y +
                leaky(a.z + b.z) * w.z + leaky(a.w + b.w) * w.w;
#pragma unroll
    for (int o = 16; o > 0; o >>= 1) sum += __shfl_xor(sum, o, 32);
    if (lane == 0) {
      score[wid * HEADS + h] = sum;
      atomicMaxFloat(&smax[d * HEADS + h], sum);
    }
  }
}

// ---------------- kernel 4: exp & segment sum -------------------------------
__global__ void gat_expsum(const long long* __restrict__ ei,
                           float* __restrict__ score,
                           const float* __restrict__ smax,
                           float* __restrict__ denom,
                           int n_edges, int n_tot) {
  int i = blockIdx.x * blockDim.x + threadIdx.x;
  if (i >= n_tot * HEADS) return;
  int e = i >> 2, h = i & 3;
  long long d = (e < n_edges) ? ei[n_edges + e] : (long long)(e - n_edges);
  float ex = __expf(score[i] - smax[d * HEADS + h]);
  score[i] = ex;                 // overwrite score with exp(score - max)
  atomAddF32(&denom[d * HEADS + h], ex);
}

// ---------------- kernel 5: alpha-weighted scatter aggregate ----------------
__global__ void gat_aggregate(const long long* __restrict__ ei,
                              const float* __restrict__ xl,
                              const float* __restrict__ score,
                              const float* __restrict__ denom,
                              float* __restrict__ out,
                              int n_edges, int n_tot) {
  const int wid  = (blockIdx.x * blockDim.x + threadIdx.x) >> 5;
  const int lane = threadIdx.x & 31;
  if (wid >= n_tot) return;

  long long s, d;
  if (wid < n_edges) { s = ei[wid]; d = ei[n_edges + wid]; }
  else               { s = d = (long long)(wid - n_edges); }

  const float* __restrict__ pl = xl + s * HC;
  float* __restrict__ po = out + d * HC;

#pragma unroll
  for (int h = 0; h < HEADS; ++h) {
    float alpha = score[wid * HEADS + h] /
                  (denom[d * HEADS + h] + 1e-16f);
    const int off = h * OUT_CH + lane * 4;
    float4 a = *(const float4*)(pl + off);
    atomAddF32(po + off + 0, alpha * a.x);
    atomAddF32(po + off + 1, alpha * a.y);
    atomAddF32(po + off + 2, alpha * a.z);
    atomAddF32(po + off + 3, alpha * a.w);
  }
}

// ---------------- host launch ----------------
extern "C" void kernel_launch(void* const* d_in, const int* in_sizes, int n_in,
                              void* d_out, int out_size, void* d_ws, size_t ws_size,
                              hipStream_t stream) {
  const float*     edge_feat = (const float*)d_in[0];       // [N,16]
  const long long* edge_index = (const long long*)d_in[1];  // [2,E] int64
  const float*     W_l  = (const float*)d_in[2];            // [16,512]
  const float*     W_r  = (const float*)d_in[3];            // [16,512]
  const float*     att  = (const float*)d_in[4];            // [4,128] flat 512
  const float*     bias = (const float*)d_in[5];            // [512]
  float* out = (float*)d_out;                               // [N,512]

  const int n_nodes = in_sizes[0] / IN_CH;                  // 50000
  const int n_edges = in_sizes[1] / 2;                      // 800000
  const int n_tot   = n_edges + n_nodes;                    // + self loops

  // workspace layout (floats)
  float* xl    = (float*)d_ws;                              // N*512
  float* xr    = xl + (size_t)n_nodes * HC;                 // N*512
  float* score = xr + (size_t)n_nodes * HC;                 // n_tot*4
  float* smax  = score + (size_t)n_tot * HEADS;             // N*4
  float* denom = smax + (size_t)n_nodes * HEADS;            // N*4

  // 1) init out = bias, smax = -inf, denom = 0
  {
    int total = n_nodes * HC;
    int blocks = (total + 255) / 256;
    gat_init<<<blocks, 256, 0, stream>>>(out, bias, smax, denom, n_nodes);
  }

  // 2) x_l / x_r GEMM via f32 WMMA. 50000 = 3125*16, 512 = 32*16 exactly.
  {
    dim3 grid(n_nodes / 16, (HC / 16) / 4, 2);   // (3125, 8, 2)
    gat_gemm_wmma<<<grid, 128, 0, stream>>>(edge_feat, W_l, W_r, xl, xr);
  }

  // 3) per-edge score + segment max (one wave per edge)
  {
    int waves_per_block = 256 / 32;
    int blocks = (n_tot + waves_per_block - 1) / waves_per_block;
    gat_score_max<<<blocks, 256, 0, stream>>>(edge_index, xl, xr, att,
                                              score, smax, n_edges, n_tot);
  }

  // 4) exp + segment sum (one thread per edge*head)
  {
    int total = n_tot * HEADS;
    int blocks = (total + 255) / 256;
    gat_expsum<<<blocks, 256, 0, stream>>>(edge_index, score, smax, denom,
                                           n_edges, n_tot);
  }

  // 5) scatter aggregate (one wave per edge)
  {
    int waves_per_block = 256 / 32;
    int blocks = (n_tot + waves_per_block - 1) / waves_per_block;
    gat_aggregate<<<blocks, 256, 0, stream>>>(edge_index, xl, score, denom,
                                              out, n_edges, n_tot);
  }
}